// GCNVirtualNode_7705171329697
// MI455X (gfx1250) — compile-verified
//
#include <hip/hip_runtime.h>

// ---------------- problem constants ----------------
static constexpr int N_NODES = 50000;
static constexpr int N_EDGES = 800000;
static constexpr int G_GRAPH = 128;
static constexpr int H_DIM   = 128;
static constexpr int OUT_DIM = 10;
static constexpr int L_LAYER = 4;
static constexpr float BN_EPS = 1e-5f;

typedef __attribute__((ext_vector_type(2))) float v2f;
typedef __attribute__((ext_vector_type(8))) float v8f;

// =====================================================================
// WMMA GEMM:  Y[N,128] = X[N,128] @ W[128,128] (+ bias)
// fp32 via V_WMMA_F32_16X16X4_F32.
// Block = 256 threads = 8 waves; wave owns a 16-row strip x 8 col tiles.
// W staged in LDS K-pair interleaved so each B fragment is ONE ds_load_b64
// straight into an even-aligned VGPR pair (no v_mov repacking).
// A fragments are unconditional global b64 loads from a clamped row:
// out-of-range rows yield garbage that only lands in never-stored D rows.
// =====================================================================
__global__ __launch_bounds__(256) void wmma_gemm128_kernel(
    const float* __restrict__ X, const float* __restrict__ W,
    const float* __restrict__ bias, float* __restrict__ Y,
    int nRows, int hasBias)
{
    // element (k, col) -> Wp[(k>>1)*256 + col*2 + (k&1)]   (64 KB)
    __shared__ float Wp[64 * 256];

    const int tid  = threadIdx.x;
    const int wave = tid >> 5;
    const int lane = tid & 31;

    for (int i = tid; i < 128 * 128; i += 256) {
        const int k = i >> 7, col = i & 127;
        Wp[(k >> 1) * 256 + col * 2 + (k & 1)] = W[i];
    }
    __syncthreads();

    const int rowBase = blockIdx.x * 128 + wave * 16;
    int mrow = rowBase + (lane & 15);
    if (mrow > nRows - 1) mrow = nRows - 1;          // clamp: always legal address
    // lanes 16..31 carry K+2 (ISA A layout): fold the +2 into the pointer
    const float* xrow = X + (size_t)mrow * 128 + ((lane >> 4) << 1);

    const int colBase = (lane & 15) * 2;             // *2: pair interleave
    const int pOff    = (lane >> 4);                 // pair row offset for hi lanes

    v8f acc[8];
#pragma unroll
    for (int ct = 0; ct < 8; ++ct) acc[ct] = (v8f)0.0f;

#pragma unroll 4
    for (int kk = 0; kk < 32; ++kk) {
        const v2f a = *(const v2f*)(xrow + kk * 4);          // one global b64
        const float* wrow = &Wp[(size_t)(kk * 2 + pOff) * 256];
#pragma unroll
        for (int ct = 0; ct < 8; ++ct) {
            const v2f b = *(const v2f*)(wrow + ct * 32 + colBase);  // one ds b64
            acc[ct] = __builtin_amdgcn_wmma_f32_16x16x4_f32(
                false, a, false, b, (short)0, acc[ct], false, false);
        }
    }

    // D layout: VGPR j -> M = j + (lane>=16 ? 8 : 0), N = lane&15
    const int srow = rowBase + (lane >> 4) * 8;
    const int scol = lane & 15;
    if (rowBase + 15 < nRows) {                       // fast path: no guards
#pragma unroll
        for (int ct = 0; ct < 8; ++ct) {
            const int col = ct * 16 + scol;
            const float bv = hasBias ? bias[col] : 0.0f;
#pragma unroll
            for (int j = 0; j < 8; ++j)
                Y[(size_t)(srow + j) * 128 + col] = acc[ct][j] + bv;
        }
    } else {
#pragma unroll
        for (int ct = 0; ct < 8; ++ct) {
            const int col = ct * 16 + scol;
            const float bv = hasBias ? bias[col] : 0.0f;
#pragma unroll
            for (int j = 0; j < 8; ++j) {
                const int r = srow + j;
                if (r < nRows) Y[(size_t)r * 128 + col] = acc[ct][j] + bv;
            }
        }
    }
}

// =====================================================================
// degree / dinv / graph counts
// =====================================================================
__global__ void deg_kernel(const int* __restrict__ dst, float* __restrict__ deg, int nE)
{
    int e = blockIdx.x * 256 + threadIdx.x;
    if (e < nE) atomicAdd(&deg[dst[e]], 1.0f);
}

__global__ void dinv_kernel(float* __restrict__ deg_dinv, int nN)
{
    int n = blockIdx.x * 256 + threadIdx.x;
    if (n < nN) deg_dinv[n] = rsqrtf(deg_dinv[n] + 1.0f);  // + self loop
}

__global__ void gcount_kernel(const int* __restrict__ batch, float* __restrict__ cnt, int nN)
{
    int n = blockIdx.x * 256 + threadIdx.x;
    if (n < nN) atomicAdd(&cnt[batch[n]], 1.0f);
}

__global__ void vn_init_kernel(const float* __restrict__ emb, float* __restrict__ vn)
{
    int t = blockIdx.x * 256 + threadIdx.x;
    if (t < G_GRAPH * H_DIM) vn[t] = emb[t & (H_DIM - 1)];
}

// =====================================================================
// edge scatter: one wave per edge; lane handles 4 channels (512B/wave)
// =====================================================================
__global__ __launch_bounds__(256) void edge_scatter_kernel(
    const float* __restrict__ h, const int* __restrict__ src,
    const int* __restrict__ dst, const float* __restrict__ dinv,
    float* __restrict__ agg, int nE)
{
    const int e = blockIdx.x * 8 + (threadIdx.x >> 5);
    if (e >= nE) return;
    const int lane = threadIdx.x & 31;
    const int s = src[e], d = dst[e];
    const float coeff = dinv[s] * dinv[d];
    const float4 v = ((const float4*)(h + (size_t)s * 128))[lane];
    float* ad = agg + (size_t)d * 128 + lane * 4;
    atomicAdd(ad + 0, v.x * coeff);
    atomicAdd(ad + 1, v.y * coeff);
    atomicAdd(ad + 2, v.z * coeff);
    atomicAdd(ad + 3, v.w * coeff);
}

// =====================================================================
// node update: self-loop + conv bias + BN + ReLU + residual (+VN) + pool
// one thread per (node, 4 channels)
// =====================================================================
__global__ __launch_bounds__(256) void node_update_kernel(
    float* __restrict__ x, const float* __restrict__ h,
    const float* __restrict__ agg, const float* __restrict__ dinv,
    const int* __restrict__ batch,
    const float* __restrict__ cb,   const float* __restrict__ g,
    const float* __restrict__ bb,   const float* __restrict__ m,
    const float* __restrict__ vvar, const float* __restrict__ vn,
    float* __restrict__ pooled, int nN, int addVN)
{
    const int idx = blockIdx.x * 256 + threadIdx.x;
    if (idx >= nN * 32) return;
    const int n  = idx >> 5;
    const int c4 = (idx & 31) * 4;
    const float di = dinv[n];
    const float d2 = di * di;
    const int b = batch[n];
    const size_t base = (size_t)n * 128 + c4;

    const float4 hv = *(const float4*)(h + base);
    const float4 av = *(const float4*)(agg + base);
    const float4 xv = *(const float4*)(x + base);
    float hin[4] = {hv.x, hv.y, hv.z, hv.w};
    float ain[4] = {av.x, av.y, av.z, av.w};
    float xin[4] = {xv.x, xv.y, xv.z, xv.w};
    float res[4];
#pragma unroll
    for (int k = 0; k < 4; ++k) {
        const int c = c4 + k;
        float t = ain[k] + hin[k] * d2 + cb[c];
        t = (t - m[c]) * rsqrtf(vvar[c] + BN_EPS) * g[c] + bb[c];
        t = fmaxf(t, 0.0f);
        float xn = xin[k] + t;
        if (addVN) xn += vn[b * 128 + c];
        res[k] = xn;
    }
    float4 o; o.x = res[0]; o.y = res[1]; o.z = res[2]; o.w = res[3];
    *(float4*)(x + base) = o;
    atomicAdd(&pooled[b * 128 + c4 + 0], res[0]);
    atomicAdd(&pooled[b * 128 + c4 + 1], res[1]);
    atomicAdd(&pooled[b * 128 + c4 + 2], res[2]);
    atomicAdd(&pooled[b * 128 + c4 + 3], res[3]);
}

// =====================================================================
// small row MLP: out[g][c] = relu(BN(in_mean[g] @ W + b))  (G=128, H=128)
// 2 graphs per 256-thread block, input rows staged in LDS.
// =====================================================================
__global__ __launch_bounds__(256) void row_mlp_kernel(
    const float* __restrict__ in, const float* __restrict__ cnt,
    const float* __restrict__ W,  const float* __restrict__ bias,
    const float* __restrict__ g,  const float* __restrict__ bb,
    const float* __restrict__ m,  const float* __restrict__ v,
    float* __restrict__ outbuf, int doBN, int useCnt)
{
    __shared__ float p[2][128];
    const int tid = threadIdx.x;
    const int g0  = blockIdx.x * 2;
    {
        const int gg = tid >> 7, k = tid & 127;
        float val = in[(g0 + gg) * 128 + k];
        if (useCnt) val /= fmaxf(cnt[g0 + gg], 1.0f);
        p[gg][k] = val;
    }
    __syncthreads();
    const int gg = tid >> 7, c = tid & 127;
    float s = bias[c];
#pragma unroll 8
    for (int k = 0; k < 128; ++k) s += p[gg][k] * W[k * 128 + c];
    if (doBN) s = (s - m[c]) * rsqrtf(v[c] + BN_EPS) * g[c] + bb[c];
    outbuf[(g0 + gg) * 128 + c] = fmaxf(s, 0.0f);
}

// vn = a*vn + (1-a)*relu(BN(h1 @ W2 + b2)),  a = sigmoid(res_w[layer])
__global__ __launch_bounds__(256) void vn_update_kernel(
    const float* __restrict__ h1, const float* __restrict__ W,
    const float* __restrict__ bias, const float* __restrict__ g,
    const float* __restrict__ bb, const float* __restrict__ m,
    const float* __restrict__ v, const float* __restrict__ res_w,
    int layer, float* __restrict__ vn)
{
    __shared__ float p[2][128];
    const int tid = threadIdx.x;
    const int g0  = blockIdx.x * 2;
    {
        const int gg = tid >> 7, k = tid & 127;
        p[gg][k] = h1[(g0 + gg) * 128 + k];
    }
    __syncthreads();
    const int gg = tid >> 7, c = tid & 127;
    float s = bias[c];
#pragma unroll 8
    for (int k = 0; k < 128; ++k) s += p[gg][k] * W[k * 128 + c];
    s = (s - m[c]) * rsqrtf(v[c] + BN_EPS) * g[c] + bb[c];
    s = fmaxf(s, 0.0f);
    const float a = 1.0f / (1.0f + __expf(-res_w[layer]));
    const int o = (g0 + gg) * 128 + c;
    vn[o] = a * vn[o] + (1.0f - a) * s;
}

// out[g][o] = hh[g] @ W2 + b2   (tiny: 128x10)
__global__ void head_out_kernel(
    const float* __restrict__ hh, const float* __restrict__ W,
    const float* __restrict__ b, float* __restrict__ out)
{
    const int t = blockIdx.x * 256 + threadIdx.x;
    if (t >= G_GRAPH * OUT_DIM) return;
    const int g = t / OUT_DIM, o = t % OUT_DIM;
    float s = b[o];
    const float* hr = hh + g * H_DIM;
#pragma unroll 8
    for (int k = 0; k < H_DIM; ++k) s += hr[k] * W[k * OUT_DIM + o];
    out[t] = s;
}

// =====================================================================
// host-side orchestration (graph-capture safe: only launches + memsetAsync)
// =====================================================================
extern "C" void kernel_launch(void* const* d_in, const int* in_sizes, int n_in,
                              void* d_out, int out_size, void* d_ws, size_t ws_size,
                              hipStream_t stream) {
    const float* x_in   = (const float*)d_in[0];
    const int*   src    = (const int*)  d_in[1];
    const int*   dst    = (const int*)  d_in[2];
    const int*   batch  = (const int*)  d_in[3];
    const float* W_in   = (const float*)d_in[4];
    const float* b_in   = (const float*)d_in[5];
    const float* conv_W = (const float*)d_in[6];
    const float* conv_b = (const float*)d_in[7];
    const float* bn_g   = (const float*)d_in[8];
    const float* bn_b   = (const float*)d_in[9];
    const float* bn_m   = (const float*)d_in[10];
    const float* bn_v   = (const float*)d_in[11];
    const float* vn_emb = (const float*)d_in[12];
    const float* vn_rw  = (const float*)d_in[13];
    const float* vn_W1  = (const float*)d_in[14];
    const float* vn_b1  = (const float*)d_in[15];
    const float* vn_g1  = (const float*)d_in[16];
    const float* vn_be1 = (const float*)d_in[17];
    const float* vn_m1  = (const float*)d_in[18];
    const float* vn_v1  = (const float*)d_in[19];
    const float* vn_W2  = (const float*)d_in[20];
    const float* vn_b2  = (const float*)d_in[21];
    const float* vn_g2  = (const float*)d_in[22];
    const float* vn_be2 = (const float*)d_in[23];
    const float* vn_m2  = (const float*)d_in[24];
    const float* vn_v2  = (const float*)d_in[25];
    const float* hd_W1  = (const float*)d_in[26];
    const float* hd_b1  = (const float*)d_in[27];
    const float* hd_W2  = (const float*)d_in[28];
    const float* hd_b2  = (const float*)d_in[29];
    float* out = (float*)d_out;

    // -------- workspace carving (256-float aligned) --------
    float* ws = (float*)d_ws;
    size_t off = 0;
    auto carve = [&](size_t n) {
        float* p = ws + off;
        off += (n + 255) & ~(size_t)255;
        return p;
    };
    float* x      = carve((size_t)N_NODES * H_DIM);
    float* h      = carve((size_t)N_NODES * H_DIM);
    float* agg    = carve((size_t)N_NODES * H_DIM);
    float* dinv   = carve(N_NODES);          // holds deg first, then rsqrt in place
    float* gcount = carve(G_GRAPH);
    float* vn     = carve(G_GRAPH * H_DIM);
    float* pooled = carve(G_GRAPH * H_DIM);
    float* h1     = carve(G_GRAPH * H_DIM);

    const int gemmGrid = (N_NODES + 127) / 128;           // 391
    const int edgeGrid = (N_EDGES + 7) / 8;               // 100000
    const int nodeGrid = (N_NODES * 32 + 255) / 256;      // 6250

    // -------- degree / dinv / per-graph counts / vn init --------
    hipMemsetAsync(dinv, 0, (size_t)N_NODES * sizeof(float), stream);
    hipMemsetAsync(gcount, 0, (size_t)G_GRAPH * sizeof(float), stream);
    deg_kernel<<<(N_EDGES + 255) / 256, 256, 0, stream>>>(dst, dinv, N_EDGES);
    dinv_kernel<<<(N_NODES + 255) / 256, 256, 0, stream>>>(dinv, N_NODES);
    gcount_kernel<<<(N_NODES + 255) / 256, 256, 0, stream>>>(batch, gcount, N_NODES);
    vn_init_kernel<<<(G_GRAPH * H_DIM + 255) / 256, 256, 0, stream>>>(vn_emb, vn);

    // -------- input projection: x = x_in @ W_in + b_in --------
    wmma_gemm128_kernel<<<gemmGrid, 256, 0, stream>>>(x_in, W_in, b_in, x, N_NODES, 1);

    // -------- GCN layers --------
    for (int i = 0; i < L_LAYER; ++i) {
        // h = x @ conv_W[i]
        wmma_gemm128_kernel<<<gemmGrid, 256, 0, stream>>>(
            x, conv_W + (size_t)i * H_DIM * H_DIM, nullptr, h, N_NODES, 0);

        hipMemsetAsync(agg, 0, (size_t)N_NODES * H_DIM * sizeof(float), stream);
        hipMemsetAsync(pooled, 0, (size_t)G_GRAPH * H_DIM * sizeof(float), stream);

        edge_scatter_kernel<<<edgeGrid, 256, 0, stream>>>(h, src, dst, dinv, agg, N_EDGES);

        const int addVN = (i < L_LAYER - 1) ? 1 : 0;
        node_update_kernel<<<nodeGrid, 256, 0, stream>>>(
            x, h, agg, dinv, batch,
            conv_b + i * H_DIM, bn_g + i * H_DIM, bn_b + i * H_DIM,
            bn_m + i * H_DIM, bn_v + i * H_DIM,
            vn, pooled, N_NODES, addVN);

        if (i < L_LAYER - 1) {
            // h1 = relu(BN(mean_pool @ vn_W1[i] + b1))
            row_mlp_kernel<<<G_GRAPH / 2, 256, 0, stream>>>(
                pooled, gcount, vn_W1 + (size_t)i * H_DIM * H_DIM, vn_b1 + i * H_DIM,
                vn_g1 + i * H_DIM, vn_be1 + i * H_DIM, vn_m1 + i * H_DIM, vn_v1 + i * H_DIM,
                h1, 1, 1);
            // vn = a*vn + (1-a)*relu(BN(h1 @ vn_W2[i] + b2))
            vn_update_kernel<<<G_GRAPH / 2, 256, 0, stream>>>(
                h1, vn_W2 + (size_t)i * H_DIM * H_DIM, vn_b2 + i * H_DIM,
                vn_g2 + i * H_DIM, vn_be2 + i * H_DIM, vn_m2 + i * H_DIM, vn_v2 + i * H_DIM,
                vn_rw, i, vn);
        }
    }

    // -------- head: relu(mean_pool @ W1 + b1) @ W2 + b2 --------
    row_mlp_kernel<<<G_GRAPH / 2, 256, 0, stream>>>(
        pooled, gcount, hd_W1, hd_b1,
        nullptr, nullptr, nullptr, nullptr, h1, 0, 1);
    head_out_kernel<<<(G_GRAPH * OUT_DIM + 255) / 256, 256, 0, stream>>>(
        h1, hd_W2, hd_b2, out);
}